// ConcentrationLoss_6665789243960
// MI455X (gfx1250) — compile-verified
//
#include <hip/hip_runtime.h>

typedef __attribute__((ext_vector_type(4))) float f32x4;
typedef __attribute__((ext_vector_type(2))) float v2f;
typedef __attribute__((ext_vector_type(8))) float v8f;

#define SPLIT 8          // blocks per (b,c) slice
#define NSLICE 160       // 16 * 10
#define HW 262144        // 512 * 512
#define F4_PER_SLICE 65536
#define F4_PER_BLOCK 8192   // 65536 / SPLIT
#define ITERS 32            // 8192 / 256 threads

__device__ __forceinline__ float fast_sigmoid(float p) {
    // 1 / (1 + e^-p); rcp(inf)=0 handles large negative p correctly.
    return __builtin_amdgcn_rcpf(1.0f + __expf(-p));
}

__device__ __forceinline__ float readlane_f(float v, int l) {
    return __int_as_float(__builtin_amdgcn_readlane(__float_as_int(v), l));
}

__global__ __launch_bounds__(256) void moments_kernel(
    const float* __restrict__ pred,
    const float* __restrict__ target,
    float* __restrict__ partial)      // [NSLICE][SPLIT][8]
{
    const int slice = blockIdx.y;                 // 0..159
    const int chunk = blockIdx.x;                 // 0..SPLIT-1
    const int tid   = threadIdx.x;
    const long sliceOff4 = (long)slice * F4_PER_SLICE;
    const f32x4* p4 = (const f32x4*)pred   + sliceOff4;
    const f32x4* t4 = (const f32x4*)target + sliceOff4;

    float T0 = 0.f, Ty = 0.f, Tx = 0.f;
    float S0 = 0.f, Sy = 0.f, Sx = 0.f, Sy2 = 0.f, Sx2 = 0.f;

    #pragma unroll 4
    for (int i = 0; i < ITERS; ++i) {
        const int fi = chunk * F4_PER_BLOCK + i * 256 + tid;   // coalesced b128
        const f32x4 p = __builtin_nontemporal_load(&p4[fi]);
        const f32x4 t = __builtin_nontemporal_load(&t4[fi]);

        const int e0 = fi << 2;                    // element index in slice
        const float y  = (float)(e0 >> 9);         // W = 512; float4 never crosses a row
        const float x0 = (float)(e0 & 511);

        const float s0 = fast_sigmoid(p.x);
        const float s1 = fast_sigmoid(p.y);
        const float s2 = fast_sigmoid(p.z);
        const float s3 = fast_sigmoid(p.w);

        const float tsum = (t.x + t.y) + (t.z + t.w);
        const float tdot = t.y + 2.f * t.z + 3.f * t.w;      // Σ t_j * j
        const float ssum = (s0 + s1) + (s2 + s3);
        const float sdot = s1 + 2.f * s2 + 3.f * s3;         // Σ s_j * j
        const float sdd  = s1 + 4.f * s2 + 9.f * s3;         // Σ s_j * j²

        T0 += tsum;
        Ty  = fmaf(y, tsum, Ty);
        Tx += fmaf(x0, tsum, tdot);
        S0 += ssum;
        Sy  = fmaf(y, ssum, Sy);
        Sy2 = fmaf(y * y, ssum, Sy2);
        Sx += fmaf(x0, ssum, sdot);
        Sx2 += fmaf(x0 * x0, ssum, fmaf(2.f * x0, sdot, sdd));
    }

    // ---- block reduction done entirely by the matrix pipe ----
    // Each thread parks its 8 moments in LDS: lds[tid*8 + j].
    // 2048 partials = 32 A-matrices (16x4 f32). Wave 0 chains 32
    // V_WMMA_F32_16X16X4_F32 ops (D accumulates into C) against an all-ones B:
    //   D[m,n] += sum_k A_i[m,k]
    // A_i[m,k] maps moment j = m&7 to rows {j, j+8}, so after the chain
    // moment j's block total = D[row j] + D[row j+8].
    __shared__ float lds[256 * 8];
    {
        f32x4 lo = {T0, Ty, Tx, S0};
        f32x4 hi = {Sy, Sx, Sy2, Sx2};
        f32x4* dst = (f32x4*)&lds[tid * 8];
        dst[0] = lo;
        dst[1] = hi;
    }
    __syncthreads();

    if (tid < 32) {   // wave 0 fully active -> EXEC all ones (WMMA requirement)
        const int lane = tid;
        const int j    = lane & 7;                   // moment (row m & 7)
        const int half = (lane >> 3) & 1;            // row m >> 3
        const int kb   = (lane >= 16) ? 2 : 0;       // A 16x4 f32 VGPR k-base
        const int baseIdx = (half * 4 + kb) * 8 + j; // thread-in-group * 8 + moment

        const v2f ones = {1.0f, 1.0f};
        v8f c = {};
        #pragma unroll
        for (int i = 0; i < 32; ++i) {
            const float* src = &lds[i * 64 + baseIdx];
            v2f a;
            a.x = src[0];                            // A[m, kb]   (thread t)
            a.y = src[8];                            // A[m, kb+1] (thread t+1)
            c = __builtin_amdgcn_wmma_f32_16x16x4_f32(
                false, a, false, ones, (short)0, c, false, false);
        }

        // total_j = D[row j] (lanes 0-15) + D[row j+8] (lanes 16-31)
        float dv[8], tot[8];
        #pragma unroll
        for (int jj = 0; jj < 8; ++jj) dv[jj] = c[jj];
        #pragma unroll
        for (int jj = 0; jj < 8; ++jj)
            tot[jj] = readlane_f(dv[jj], 0) + readlane_f(dv[jj], 16);

        if (lane == 0) {
            float* outp = partial + ((long)(slice * SPLIT + chunk) * 8);
            f32x4 o0 = {tot[0], tot[1], tot[2], tot[3]};
            f32x4 o1 = {tot[4], tot[5], tot[6], tot[7]};
            *(f32x4*)(outp)     = o0;
            *(f32x4*)(outp + 4) = o1;
        }
    }
}

__global__ __launch_bounds__(256) void finalize_kernel(
    const float* __restrict__ partial, float* __restrict__ out)
{
    __shared__ float ssum[256];
    __shared__ float scnt[256];
    const int tid = threadIdx.x;

    float loss = 0.f, cnt = 0.f;
    if (tid < NSLICE) {
        float m[8] = {0.f, 0.f, 0.f, 0.f, 0.f, 0.f, 0.f, 0.f};
        for (int k = 0; k < SPLIT; ++k) {            // fixed order -> deterministic
            const float* p = partial + ((long)(tid * SPLIT + k) * 8);
            #pragma unroll
            for (int j = 0; j < 8; ++j) m[j] += p[j];
        }
        const float T0 = m[0], Ty = m[1], Tx = m[2];
        const float S0 = m[3], Sy = m[4], Sx = m[5], Sy2 = m[6], Sx2 = m[7];
        const bool valid = T0 > 0.f;
        const float safe = valid ? T0 : 1.f;
        const float cy = Ty / safe;
        const float cx = Tx / safe;
        // Σ s·dist² = Sy2 + Sx2 + S0·(cy²+cx²) − 2·(cy·Sy + cx·Sx)
        const float sumVal = Sy2 + Sx2 + S0 * (cy * cy + cx * cx)
                           - 2.f * (cy * Sy + cx * Sx);
        const float per = sumVal * (1.0f / (float)HW);
        loss = valid ? per : 0.f;
        cnt  = valid ? 1.f : 0.f;
    }
    ssum[tid] = loss;
    scnt[tid] = cnt;
    __syncthreads();
    for (int s = 128; s > 0; s >>= 1) {
        if (tid < s) { ssum[tid] += ssum[tid + s]; scnt[tid] += scnt[tid + s]; }
        __syncthreads();
    }
    if (tid == 0) out[0] = (scnt[0] > 0.f) ? (ssum[0] / scnt[0]) : 0.f;
}

extern "C" void kernel_launch(void* const* d_in, const int* in_sizes, int n_in,
                              void* d_out, int out_size, void* d_ws, size_t ws_size,
                              hipStream_t stream) {
    const float* pred   = (const float*)d_in[0];
    const float* target = (const float*)d_in[1];
    float* out = (float*)d_out;
    float* partial = (float*)d_ws;   // needs NSLICE*SPLIT*8*4 = 40960 bytes

    dim3 grid(SPLIT, NSLICE);        // 1280 blocks x 8 waves = 10240 waves
    moments_kernel<<<grid, 256, 0, stream>>>(pred, target, partial);
    finalize_kernel<<<1, 256, 0, stream>>>(partial, out);
}